// MultiHeadAttention_44890998178158
// MI455X (gfx1250) — compile-verified
//
#include <hip/hip_runtime.h>
#include <math.h>

// MI455X / gfx1250: wave32, fp32 WMMA 16x16x4, 320KB-LDS-resident softmax.
// Problem: B=2, L=2048, D_MODEL=256, H=8, D_HEAD=32.
// d_out = [out(B,L,256) | attn(B,H,L,L)] concatenated, fp32.
// d_ws layout (floats): Q[1M] | K[1M] | V[1M] | Ctx[1M]  == 16 MB total.

typedef __attribute__((ext_vector_type(2))) float v2f;
typedef __attribute__((ext_vector_type(8))) float v8f;

#define B_   2
#define L_   2048
#define DM_  256
#define H_   8
#define DH_  32
#define INV_SCALE 0.17677669529663687f   // 1/sqrt(D_HEAD)

#define SSTRIDE 2052                     // LDS stripe row stride (pad: 2052%64==4)
#define NKT     (L_ / 16)                // 128 k-tiles

static __device__ __forceinline__ v8f wmma_f32(v2f a, v2f b, v8f c) {
    // D = A(16x4 f32) * B(4x16 f32) + C(16x16 f32)
    return __builtin_amdgcn_wmma_f32_16x16x4_f32(false, a, false, b, (short)0, c,
                                                 false, false);
}

// ---------------------------------------------------------------------------
// Y = X @ W^T for the three input projections, stored as [B,H,L,DH].
// One wave per 16x16 output tile; 64 chained fp32 WMMAs.
// ---------------------------------------------------------------------------
__global__ void proj_qkv_kernel(const float* __restrict__ Xq,
                                const float* __restrict__ Xk,
                                const float* __restrict__ Xv,
                                const float* __restrict__ Wq,
                                const float* __restrict__ Wk,
                                const float* __restrict__ Wv,
                                float* __restrict__ Qo,
                                float* __restrict__ Ko,
                                float* __restrict__ Vo) {
    const int lane = threadIdx.x;
    const int half = lane >> 4;
    const int l16  = lane & 15;
    const int jt   = blockIdx.x;     // col tile (0..15)
    const int it   = blockIdx.y;     // row tile (0..255) over B*L
    const int sel  = blockIdx.z;

    const float* X = (sel == 0) ? Xq : (sel == 1) ? Xk : Xv;
    const float* W = (sel == 0) ? Wq : (sel == 1) ? Wk : Wv;
    float*       Y = (sel == 0) ? Qo : (sel == 1) ? Ko : Vo;

    const float* xrow = X + (size_t)(it * 16 + l16) * DM_;  // A: M = l16
    const float* wrow = W + (size_t)(jt * 16 + l16) * DM_;  // B: N = l16

    v8f acc = {};
#pragma unroll
    for (int c = 0; c < DM_ / 4; ++c) {
        const int d = 4 * c + 2 * half;
        v2f a = *(const v2f*)(xrow + d);
        v2f b = *(const v2f*)(wrow + d);
        acc = wmma_f32(a, b, acc);
    }

    const int h  = jt >> 1;
    const int dh = ((jt & 1) << 4) + l16;
#pragma unroll
    for (int r = 0; r < 8; ++r) {
        const int m = it * 16 + r + 8 * half;
        const int b = m / L_;
        const int l = m - b * L_;
        Y[(((size_t)b * H_ + h) * L_ + l) * DH_ + dh] = acc[r];
    }
}

// ---------------------------------------------------------------------------
// Attention. Workgroup = 4 waves = one (b, h, 16 query rows) block.
// LDS (dynamic, ~134KB): score stripe 16 x L (stride SSTRIDE) + reduce scratch.
//   sweep1 : waves split k-tiles 4-way. S=QK^T (WMMA), modulate, store->LDS,
//            partial rowmax.
//   sweep2 : rowsum of exp from LDS stripe.
//   phaseA : p = exp(s-max)/sum; LDS in-place; single NT stream to attn out.
//   phaseB : O = P@V; P read straight from LDS stripe in WMMA-A layout;
//            waves = 2 d-blocks x 2 k-halves, O combined through LDS.
// grid = B*H*(L/16) = 2048, block = 128.
// ---------------------------------------------------------------------------
__global__ void attn_kernel(const float* __restrict__ Q,     // [B,H,L,DH]
                            const float* __restrict__ K,
                            const float* __restrict__ V,
                            const int*   __restrict__ mask,  // [B,L,L]
                            const float* __restrict__ adj,   // [B,L,L]
                            const float* __restrict__ dist,  // [B,L,L]
                            const float* __restrict__ wA,    // [H]
                            const float* __restrict__ wD,    // [H]
                            float* __restrict__ attn,        // [B,H,L,L]
                            float* __restrict__ ctx) {       // [B,L,DM]
    extern __shared__ float smem[];
    float* sS     = smem;                    // [16][SSTRIDE] score stripe
    float* redmax = smem + 16 * SSTRIDE;     // [4][16]
    float* redsum = redmax + 64;             // [4][16]
    float* ocomb  = redsum + 64;             // [2][256]

    const int tid  = threadIdx.x;
    const int w    = tid >> 5;               // wave 0..3
    const int lane = tid & 31;
    const int half = lane >> 4;
    const int l16  = lane & 15;

    const int bid = blockIdx.x;
    const int qt  = bid & 127;
    const int h   = (bid >> 7) & 7;
    const int b   = bid >> 10;
    const int qbase = qt * 16;

    const size_t bh = (size_t)b * H_ + h;
    const float* Qh = Q + bh * L_ * DH_;
    const float* Kh = K + bh * L_ * DH_;
    const float* Vh = V + bh * L_ * DH_;
    float* attn_h   = attn + bh * (size_t)L_ * L_;

    const float wa = wA[h];
    const float wd = wD[h];

    // Per-wave copy of the 16x32 Q tile in WMMA-A layout.
    v2f qa[8];
    {
        const float* qrow = Qh + (size_t)(qbase + l16) * DH_;
#pragma unroll
        for (int c = 0; c < 8; ++c) qa[c] = *(const v2f*)(qrow + 4 * c + 2 * half);
    }

    // ---- sweep 1: scores -> LDS stripe, partial rowmax ---------------------
    float rowmax[8];
#pragma unroll
    for (int r = 0; r < 8; ++r) rowmax[r] = -INFINITY;

    for (int kt = w; kt < NKT; kt += 4) {
        const int kbase = kt * 16;
        const float* krow = Kh + (size_t)(kbase + l16) * DH_;
        v8f s8 = {};
#pragma unroll
        for (int c = 0; c < 8; ++c) {
            v2f kb = *(const v2f*)(krow + 4 * c + 2 * half);
            s8 = wmma_f32(qa[c], kb, s8);
        }
        const int kcol = kbase + l16;
#pragma unroll
        for (int r = 0; r < 8; ++r) {
            const int m = r + 8 * half;                 // row within tile
            const size_t midx = ((size_t)b * L_ + qbase + m) * L_ + kcol;
            float s = s8[r];
            s = (__builtin_nontemporal_load(mask + midx) == 1) ? s : 0.0f;
            s = s * (1.0f + __builtin_nontemporal_load(adj + midx) * wa
                          + __builtin_nontemporal_load(dist + midx) * wd) * INV_SCALE;
            rowmax[r] = fmaxf(rowmax[r], s);
            sS[m * SSTRIDE + kcol] = s;
        }
    }
    // intra-wave then cross-wave rowmax
#pragma unroll
    for (int off = 1; off < 16; off <<= 1)
#pragma unroll
        for (int r = 0; r < 8; ++r)
            rowmax[r] = fmaxf(rowmax[r], __shfl_xor(rowmax[r], off, 32));
    if (l16 == 0) {
#pragma unroll
        for (int r = 0; r < 8; ++r) redmax[w * 16 + r + 8 * half] = rowmax[r];
    }
    __syncthreads();
#pragma unroll
    for (int r = 0; r < 8; ++r) {
        const int m = r + 8 * half;
        rowmax[r] = fmaxf(fmaxf(redmax[m], redmax[16 + m]),
                          fmaxf(redmax[32 + m], redmax[48 + m]));
    }

    // ---- sweep 2: rowsum of exp from LDS -----------------------------------
    float rowsum[8];
#pragma unroll
    for (int r = 0; r < 8; ++r) rowsum[r] = 0.0f;
    for (int kt = w; kt < NKT; kt += 4) {
        const int kcol = kt * 16 + l16;
#pragma unroll
        for (int r = 0; r < 8; ++r) {
            const int m = r + 8 * half;
            rowsum[r] += __expf(sS[m * SSTRIDE + kcol] - rowmax[r]);
        }
    }
#pragma unroll
    for (int off = 1; off < 16; off <<= 1)
#pragma unroll
        for (int r = 0; r < 8; ++r)
            rowsum[r] += __shfl_xor(rowsum[r], off, 32);
    if (l16 == 0) {
#pragma unroll
        for (int r = 0; r < 8; ++r) redsum[w * 16 + r + 8 * half] = rowsum[r];
    }
    __syncthreads();
    float rinv[8];
#pragma unroll
    for (int r = 0; r < 8; ++r) {
        const int m = r + 8 * half;
        rinv[r] = 1.0f / (redsum[m] + redsum[16 + m] + redsum[32 + m] + redsum[48 + m]);
    }

    // ---- phase A: normalize in LDS, single NT stream of attn to HBM --------
    for (int kt = w; kt < NKT; kt += 4) {
        const int kcol = kt * 16 + l16;
#pragma unroll
        for (int r = 0; r < 8; ++r) {
            const int m = r + 8 * half;
            const float p = __expf(sS[m * SSTRIDE + kcol] - rowmax[r]) * rinv[r];
            sS[m * SSTRIDE + kcol] = p;
            __builtin_nontemporal_store(p, attn_h + (size_t)(qbase + m) * L_ + kcol);
        }
    }
    __syncthreads();

    // ---- phase B: O = P @ V, P read from LDS stripe in A-layout ------------
    const int nb    = w & 1;    // d-block (cols nb*16 .. nb*16+15)
    const int khalf = w >> 1;   // k split
    v8f o = {};
    for (int kt = khalf; kt < NKT; kt += 2) {
        const int kbase = kt * 16;
#pragma unroll
        for (int c = 0; c < 4; ++c) {
            const int kk = 4 * c + 2 * half;
            v2f pa = *(const v2f*)(&sS[l16 * SSTRIDE + kbase + kk]);  // A: M=l16
            const float* v0 = Vh + (size_t)(kbase + kk) * DH_ + nb * 16;
            v2f bb;
            bb.x = v0[l16];
            bb.y = v0[DH_ + l16];
            o = wmma_f32(pa, bb, o);
        }
    }
    // combine the two k-halves per d-block through LDS
    if (khalf == 1) {
#pragma unroll
        for (int r = 0; r < 8; ++r) ocomb[nb * 256 + r * 32 + lane] = o[r];
    }
    __syncthreads();
    if (khalf == 0) {
#pragma unroll
        for (int r = 0; r < 8; ++r) {
            const int m = r + 8 * half;
            const float val = o[r] + ocomb[nb * 256 + r * 32 + lane];
            ctx[((size_t)b * L_ + qbase + m) * DM_ + h * DH_ + nb * 16 + l16] = val;
        }
    }
}

// ---------------------------------------------------------------------------
// out = Ctx @ Wo^T -> d_out[0 : B*L*DM]. grid = (16, 256), block = 32.
// ---------------------------------------------------------------------------
__global__ void out_proj_kernel(const float* __restrict__ X,
                                const float* __restrict__ Wo,
                                float* __restrict__ Y) {
    const int lane = threadIdx.x;
    const int half = lane >> 4;
    const int l16  = lane & 15;
    const int jt   = blockIdx.x;
    const int it   = blockIdx.y;

    const float* xrow = X  + (size_t)(it * 16 + l16) * DM_;
    const float* wrow = Wo + (size_t)(jt * 16 + l16) * DM_;

    v8f acc = {};
#pragma unroll
    for (int c = 0; c < DM_ / 4; ++c) {
        const int d = 4 * c + 2 * half;
        v2f a = *(const v2f*)(xrow + d);
        v2f b = *(const v2f*)(wrow + d);
        acc = wmma_f32(a, b, acc);
    }
#pragma unroll
    for (int r = 0; r < 8; ++r) {
        const int m = it * 16 + r + 8 * half;
        Y[(size_t)m * DM_ + jt * 16 + l16] = acc[r];
    }
}

// ---------------------------------------------------------------------------
extern "C" void kernel_launch(void* const* d_in, const int* in_sizes, int n_in,
                              void* d_out, int out_size, void* d_ws, size_t ws_size,
                              hipStream_t stream) {
    const float* queries = (const float*)d_in[0];
    const float* keys    = (const float*)d_in[1];
    const float* values  = (const float*)d_in[2];
    const int*   amask   = (const int*)  d_in[3];
    const float* adj     = (const float*)d_in[4];
    const float* dist    = (const float*)d_in[5];
    const float* Wq      = (const float*)d_in[6];
    const float* Wk      = (const float*)d_in[7];
    const float* Wv      = (const float*)d_in[8];
    const float* Wo      = (const float*)d_in[9];
    const float* wA      = (const float*)d_in[10];  // [1,H,1,1] -> 8 floats
    const float* wD      = (const float*)d_in[11];

    float* out_final = (float*)d_out;                           // [B,L,DM]
    float* attn_out  = out_final + (size_t)B_ * L_ * DM_;       // [B,H,L,L]

    const size_t qkv = (size_t)B_ * H_ * L_ * DH_;              // 1M floats
    float* ws  = (float*)d_ws;
    float* Qws = ws;
    float* Kws = ws + qkv;
    float* Vws = ws + 2 * qkv;
    float* Ctx = ws + 3 * qkv;

    // 1) Q/K/V projections via fp32 WMMA.
    proj_qkv_kernel<<<dim3(DM_ / 16, (B_ * L_) / 16, 3), 32, 0, stream>>>(
        queries, keys, values, Wq, Wk, Wv, Qws, Kws, Vws);

    // 2) Attention with LDS-resident score stripe (~134 KB dynamic LDS/WG).
    const size_t lds_bytes = (16 * SSTRIDE + 64 + 64 + 512) * sizeof(float);
    attn_kernel<<<B_ * H_ * (L_ / 16), 128, lds_bytes, stream>>>(
        Qws, Kws, Vws, amask, adj, dist, wA, wD, attn_out, Ctx);

    // 3) Output projection.
    out_proj_kernel<<<dim3(DM_ / 16, (B_ * L_) / 16), 32, 0, stream>>>(
        Ctx, Wo, out_final);
}